// TCR_Loss_43104291782750
// MI455X (gfx1250) — compile-verified
//
#include <hip/hip_runtime.h>
#include <math.h>

typedef __attribute__((ext_vector_type(2))) float v2f;
typedef __attribute__((ext_vector_type(8))) float v8f;

// Problem constants (from reference setup_inputs)
#define BB 64
#define CC 3
#define HH 256
#define WW 256
#define NIMG (BB * CC * HH * WW)       // 12,582,912 floats
#define KTOT 27                        // Cin*3*3
#define KSTEPS 7                       // ceil(27/4) K-steps of the 16x16x4 f32 WMMA
#define TILES (BB * HH * (WW / 16))    // 16-pixel tiles
#define ANG_F 0.17453292519943295f     // 10 degrees in radians
#define MAXT_F 6.0f
#define NB_CONV 512                    // blocks for WMMA kernels (256 thr = 8 waves)

// ---------------------------------------------------------------------------
// Kernel 1: per-batch affine T and its inverse (bottom row is [0,0,1])
// ---------------------------------------------------------------------------
__global__ void build_minv_kernel(const float* __restrict__ rot,
                                  const float* __restrict__ txr,
                                  const float* __restrict__ tyr,
                                  const float* __restrict__ tzr,
                                  float* __restrict__ minv) {
    int b = threadIdx.x;
    if (b >= BB) return;
    const float Wd = (float)HH;  // reference: Wd = img.shape[2] (=256)
    const float Hd = (float)WW;  // reference: Hd = img.shape[3] (=256)
    float tx = 2.0f * MAXT_F * txr[b] - MAXT_F;
    float ty = 2.0f * MAXT_F * tyr[b] - MAXT_F;
    float r  = 2.0f * ANG_F * rot[b] - ANG_F;
    float z  = 1.0f;               // (MAX_Z-MAX_Z)*rand + MAX_Z
    (void)tzr;
    float hx = r, hy = r;
    float a  = hx - r;             // 0
    float bb = hy + r;             // 2r
    float cos_hx = cosf(hx), cos_hy = cosf(hy);
    float ca = cosf(a),  sa = sinf(a);
    float cb = cosf(bb), sb = sinf(bb);
    float T11 = z * ca / cos_hx;
    float T12 = z * sa / cos_hx;
    float T13 = (Wd * cos_hx - Wd * z * ca + 2.f * tx * z * ca
                 - Hd * z * sa + 2.f * ty * z * sa) / (2.f * cos_hx);
    float T21 = z * sb / cos_hy;
    float T22 = z * cb / cos_hy;
    float T23 = (Hd * cos_hy - Wd * z * cb + 2.f * ty * z * cb
                 - Wd * z * sb + 2.f * tx * z * sb) / (2.f * cos_hy);
    float det = T11 * T22 - T12 * T21;
    float inv = 1.0f / det;
    float* M = minv + b * 6;
    M[0] =  T22 * inv;
    M[1] = -T12 * inv;
    M[2] = (T12 * T23 - T13 * T22) * inv;
    M[3] = -T21 * inv;
    M[4] =  T11 * inv;
    M[5] = (T13 * T21 - T11 * T23) * inv;
}

// ---------------------------------------------------------------------------
// Implicit-GEMM 3x3 conv helpers.  K = ci*9 + ky*3 + kx  (ky,kx in 0..2)
// ISA 16x4 f32 A/B layout: lanes 0-15 hold {K=k0,K=k0+1}, lanes 16-31 {k0+2,k0+3}.
// ---------------------------------------------------------------------------

// A-fragments (weights) for all 7 K-steps; K>=27 and rows m>=3 are zero, which
// also neutralizes any dummy B values loaded for the padded K column.
__device__ __forceinline__ void load_afrags(const float* __restrict__ w,
                                            int lane, v2f afrag[KSTEPS]) {
    int m  = lane & 15;
    int kb = (lane < 16) ? 0 : 2;
#pragma unroll
    for (int s = 0; s < KSTEPS; ++s) {
        int k0 = 4 * s + kb;
        float ax = 0.f, ay = 0.f;
        if (m < CC) {
            if (k0     < KTOT) ax = w[m * KTOT + k0];
            if (k0 + 1 < KTOT) ay = w[m * KTOT + k0 + 1];
        }
        afrag[s].x = ax;
        afrag[s].y = ay;
    }
}

// Folded per-lane tap offsets for the interior fast path:
//   foff = ci*H*W + dy*W + dx  (dy,dx in -1..1).  Invalid K -> offset 0
// (the dummy load lands in-bounds and is multiplied by a zero A column).
__device__ __forceinline__ void build_foff(int lane, int foff[KSTEPS][2]) {
    int kb = (lane < 16) ? 0 : 2;
#pragma unroll
    for (int s = 0; s < KSTEPS; ++s) {
#pragma unroll
        for (int j = 0; j < 2; ++j) {
            int k = 4 * s + kb + j;
            int o = 0;
            if (k < KTOT) {
                int ci  = k / 9;
                int rem = k - ci * 9;
                int dy  = rem / 3 - 1;
                int dx  = rem - (dy + 1) * 3 - 1;
                o = ci * HH * WW + dy * WW + dx;
            }
            foff[s][j] = o;
        }
    }
}

// Branch-free edge tap: wrap address with &255 (stays in-bounds for W=H=256),
// zero the VALUE with one unsigned compare instead of EXEC-masked loads.
__device__ __forceinline__ float conv_sample_edge(const float* __restrict__ imgb,
                                                  int k, int yrow, int x) {
    int kc  = (k < KTOT) ? k : 0;
    int ci  = kc / 9;
    int rem = kc - ci * 9;
    int dy  = rem / 3 - 1;
    int dx  = rem - (dy + 1) * 3 - 1;
    int yy = yrow + dy;
    int xx = x + dx;
    float v = imgb[(size_t)ci * (HH * WW) + ((yy & (HH - 1)) << 8) + (xx & (WW - 1))];
    bool ok = ((unsigned)(yy | xx) <= (unsigned)(WW - 1)) && (k < KTOT);
    return ok ? v : 0.f;
}

// One 16-pixel conv tile: 7 accumulating fp32 WMMAs.  Rows 0..2 of C = channels.
__device__ __forceinline__ v8f conv_tile_wmma(const float* __restrict__ imgb,
                                              const v2f afrag[KSTEPS],
                                              const int foff[KSTEPS][2],
                                              int lane, int yrow, int x0) {
    int n = lane & 15;
    int x = x0 + n;
    v8f acc = {};
    // wave-uniform interior test: taps cover rows yrow-1..yrow+1, cols x0-1..x0+16
    if (yrow >= 1 && yrow <= HH - 2 && x0 >= 16 && x0 <= WW - 32) {
        const float* p = imgb + yrow * WW + x;
#pragma unroll
        for (int s = 0; s < KSTEPS; ++s) {
            v2f bf;
            bf.x = p[foff[s][0]];
            bf.y = p[foff[s][1]];
            acc = __builtin_amdgcn_wmma_f32_16x16x4_f32(
                false, afrag[s], false, bf, (short)0, acc, false, false);
        }
    } else {
        int kb = (lane < 16) ? 0 : 2;
#pragma unroll
        for (int s = 0; s < KSTEPS; ++s) {
            v2f bf;
            bf.x = conv_sample_edge(imgb, 4 * s + kb,     yrow, x);
            bf.y = conv_sample_edge(imgb, 4 * s + kb + 1, yrow, x);
            acc = __builtin_amdgcn_wmma_f32_16x16x4_f32(
                false, afrag[s], false, bf, (short)0, acc, false, false);
        }
    }
    return acc;
}

// ---------------------------------------------------------------------------
// Kernel 2: y = conv(img) via fp32 WMMA implicit GEMM
// ---------------------------------------------------------------------------
__global__ void conv_img_wmma_kernel(const float* __restrict__ img,
                                     const float* __restrict__ w,
                                     float* __restrict__ ybuf) {
    int lane   = threadIdx.x & 31;
    int wid    = (blockIdx.x * blockDim.x + threadIdx.x) >> 5;
    int nwaves = (gridDim.x * blockDim.x) >> 5;

    v2f afrag[KSTEPS];
    load_afrags(w, lane, afrag);
    int foff[KSTEPS][2];
    build_foff(lane, foff);

    for (int t = wid; t < TILES; t += nwaves) {
        int x0   = (t & 15) * 16;
        int yrow = (t >> 4) & (HH - 1);
        int b    = t >> 12;
        const float* imgb = img + (size_t)b * CC * HH * WW;
        v8f acc = conv_tile_wmma(imgb, afrag, foff, lane, yrow, x0);
        if (lane < 16) {
            int x = x0 + lane;
#pragma unroll
            for (int ch = 0; ch < CC; ++ch)
                ybuf[(((size_t)b * CC + ch) * HH + yrow) * WW + x] = acc[ch];
        }
    }
}

// ---------------------------------------------------------------------------
// Branch-free bilinear tap (zero outside): wrap address, mask the weight.
// ---------------------------------------------------------------------------
__device__ __forceinline__ float warp_sample(const float* __restrict__ srcb,
                                             int c, float sx, float sy) {
    float x0f = floorf(sx), y0f = floorf(sy);
    float wx = sx - x0f, wy = sy - y0f;
    int x0 = (int)x0f, y0 = (int)y0f;
    const float* ch = srcb + (size_t)c * HH * WW;
    float v = 0.f;
#pragma unroll
    for (int j = 0; j < 2; ++j) {
#pragma unroll
        for (int i = 0; i < 2; ++i) {
            int xi = x0 + i, yi = y0 + j;
            float wgt = (i ? wx : 1.f - wx) * (j ? wy : 1.f - wy);
            float t = ch[((yi & (HH - 1)) << 8) + (xi & (WW - 1))];
            bool ok = (unsigned)(yi | xi) <= (unsigned)(WW - 1);
            v += ok ? t * wgt : 0.f;
        }
    }
    return v;
}

// ---------------------------------------------------------------------------
// Kernel 3: wimg = warp_affine(img, Minv)
// ---------------------------------------------------------------------------
__global__ void warp_kernel(const float* __restrict__ src,
                            const float* __restrict__ minv,
                            float* __restrict__ dst) {
    int idx = blockIdx.x * blockDim.x + threadIdx.x;
    if (idx >= BB * HH * WW) return;
    int x    = idx & (WW - 1);
    int yrow = (idx >> 8) & (HH - 1);
    int b    = idx >> 16;
    const float* M = minv + b * 6;
    float sx = M[0] * (float)x + M[1] * (float)yrow + M[2];
    float sy = M[3] * (float)x + M[4] * (float)yrow + M[5];
    const float* srcb = src + (size_t)b * CC * HH * WW;
    int ycl = min(max((int)sy, 0), HH - 1);
    __builtin_prefetch(srcb + (size_t)ycl * WW, 0, 3);
#pragma unroll
    for (int c = 0; c < CC; ++c)
        dst[(((size_t)b * CC + c) * HH + yrow) * WW + x] = warp_sample(srcb, c, sx, sy);
}

// ---------------------------------------------------------------------------
// Kernel 4: z = conv(wimg) (WMMA), wy = warp(y) on the fly, accumulate (z-wy)^2
// ---------------------------------------------------------------------------
__global__ void fused_loss_wmma_kernel(const float* __restrict__ wimg,
                                       const float* __restrict__ w,
                                       const float* __restrict__ ybuf,
                                       const float* __restrict__ minv,
                                       float* __restrict__ partials) {
    int lane   = threadIdx.x & 31;
    int wid    = (blockIdx.x * blockDim.x + threadIdx.x) >> 5;
    int nwaves = (gridDim.x * blockDim.x) >> 5;

    v2f afrag[KSTEPS];
    load_afrags(w, lane, afrag);
    int foff[KSTEPS][2];
    build_foff(lane, foff);

    float lsum = 0.f;
    for (int t = wid; t < TILES; t += nwaves) {
        int x0   = (t & 15) * 16;
        int yrow = (t >> 4) & (HH - 1);
        int b    = t >> 12;
        const float* wib = wimg + (size_t)b * CC * HH * WW;
        v8f acc = conv_tile_wmma(wib, afrag, foff, lane, yrow, x0);
        if (lane < 16) {
            int x = x0 + lane;
            const float* M = minv + b * 6;
            float sx = M[0] * (float)x + M[1] * (float)yrow + M[2];
            float sy = M[3] * (float)x + M[4] * (float)yrow + M[5];
            const float* yb = ybuf + (size_t)b * CC * HH * WW;
#pragma unroll
            for (int ch = 0; ch < CC; ++ch) {
                float d = acc[ch] - warp_sample(yb, ch, sx, sy);
                lsum += d * d;
            }
        }
    }
    __shared__ float sdata[256];
    int tid = threadIdx.x;
    sdata[tid] = lsum;
    __syncthreads();
    for (int s = 128; s > 0; s >>= 1) {
        if (tid < s) sdata[tid] += sdata[tid + s];
        __syncthreads();
    }
    if (tid == 0) partials[blockIdx.x] = sdata[0];
}

// ---------------------------------------------------------------------------
// Kernel 5: deterministic final reduction -> mean
// ---------------------------------------------------------------------------
__global__ void final_reduce_kernel(const float* __restrict__ partials,
                                    int n, float* __restrict__ out) {
    __shared__ float sdata[256];
    int tid = threadIdx.x;
    float s = 0.f;
    for (int i = tid; i < n; i += 256) s += partials[i];
    sdata[tid] = s;
    __syncthreads();
    for (int k = 128; k > 0; k >>= 1) {
        if (tid < k) sdata[tid] += sdata[tid + k];
        __syncthreads();
    }
    if (tid == 0) out[0] = sdata[0] / (float)NIMG;
}

// ---------------------------------------------------------------------------
extern "C" void kernel_launch(void* const* d_in, const int* in_sizes, int n_in,
                              void* d_out, int out_size, void* d_ws, size_t ws_size,
                              hipStream_t stream) {
    const float* img  = (const float*)d_in[0];
    const float* rot  = (const float*)d_in[1];
    const float* trx  = (const float*)d_in[2];
    const float* try_ = (const float*)d_in[3];
    const float* trz  = (const float*)d_in[4];
    const float* cw   = (const float*)d_in[5];
    float* out = (float*)d_out;
    (void)in_sizes; (void)n_in; (void)out_size; (void)ws_size;

    float* ws       = (float*)d_ws;
    float* ybuf     = ws;                    // NIMG floats: conv(img)
    float* wimg     = ws + (size_t)NIMG;     // NIMG floats: warp(img)
    float* minv     = ws + (size_t)2 * NIMG; // 64*6 floats
    float* partials = minv + 512;            // NB_CONV floats

    build_minv_kernel<<<1, 64, 0, stream>>>(rot, trx, try_, trz, minv);
    conv_img_wmma_kernel<<<NB_CONV, 256, 0, stream>>>(img, cw, ybuf);
    int nwarp = BB * HH * WW;
    warp_kernel<<<(nwarp + 255) / 256, 256, 0, stream>>>(img, minv, wimg);
    fused_loss_wmma_kernel<<<NB_CONV, 256, 0, stream>>>(wimg, cw, ybuf, minv, partials);
    final_reduce_kernel<<<1, 256, 0, stream>>>(partials, NB_CONV, out);
}